// GraphPolicyNetwork_48344151884052
// MI455X (gfx1250) — compile-verified
//
#include <hip/hip_runtime.h>

#define N_NODES 50000
#define N_EDGES 640000
#define D_IN    128
#define D_HID   256
#define D_OUT   2

// ws layout (float offsets); deg/agg1/agg2 contiguous so one zero pass covers them
#define OFF_DEG  0
#define OFF_AGG1 (50000)
#define OFF_AGG2 (50000 + 6400000)
#define OFF_H    (50000 + 6400000 + 100000)
#define OFF_Y    (OFF_H + 12800000)
#define ZERO_N   (50000 + 6400000 + 100000)

typedef __attribute__((ext_vector_type(2))) float v2f;
typedef __attribute__((ext_vector_type(8))) float v8f;

__global__ void zero_f32(float* __restrict__ p, int n) {
    int i = blockIdx.x * blockDim.x + threadIdx.x;
    if (i < n) p[i] = 0.0f;
}

__global__ void degree_kernel(const int* __restrict__ dst, float* __restrict__ deg) {
    int e = blockIdx.x * blockDim.x + threadIdx.x;
    if (e < N_EDGES) atomicAdd(&deg[dst[e]], 1.0f);
}

// lane-per-feature edge scatter: coalesced 512B gather per edge, coalesced atomics
__global__ void scatter1_kernel(const float* __restrict__ x,
                                const int* __restrict__ src,
                                const int* __restrict__ dst,
                                float* __restrict__ agg1) {
    int i = blockIdx.x * blockDim.x + threadIdx.x;   // < E*128 = 81.92M
    int e = i >> 7;
    int d = i & 127;
    if (e < N_EDGES) {
        atomicAdd(&agg1[dst[e] * D_IN + d], x[src[e] * D_IN + d]);
    }
}

// One wave computes a 16x16 tile of h = relu(x@W1s + (agg1/deg)@W1n + b1).
// Two independent WMMA accumulator chains (self / neigh) for pipelining.
__global__ void gemm1_wmma(const float* __restrict__ x,
                           const float* __restrict__ agg1,
                           const float* __restrict__ deg,
                           const float* __restrict__ W1s,
                           const float* __restrict__ W1n,
                           const float* __restrict__ b1,
                           float* __restrict__ h) {
    int wave = (blockIdx.x * blockDim.x + threadIdx.x) >> 5;  // 0..49999
    int lane = threadIdx.x & 31;
    int m = wave >> 4;      // node tile 0..3124
    int j = wave & 15;      // output-col tile 0..15
    int half = lane >> 4;   // K-split for A, M+8 split for C/D
    int l15  = lane & 15;
    int rowA = m * 16 + l15;            // A row owned by this lane (fixed over K)
    int col  = j * 16 + l15;            // B/C/D column owned by this lane
    float invd = 1.0f / fmaxf(deg[rowA], 1.0f);
    const float* xr = x    + rowA * D_IN;
    const float* gr = agg1 + rowA * D_IN;

    v8f acc_s = {0.f,0.f,0.f,0.f,0.f,0.f,0.f,0.f};
    v8f acc_n = {0.f,0.f,0.f,0.f,0.f,0.f,0.f,0.f};

#pragma unroll 4
    for (int k = 0; k < D_IN; k += 4) {
        int ka = k + half * 2;
        v2f a0, b0, a1, b1v;
        a0.x = xr[ka];              a0.y = xr[ka + 1];
        b0.x = W1s[ka * D_HID + col];
        b0.y = W1s[(ka + 1) * D_HID + col];
        a1.x = gr[ka] * invd;       a1.y = gr[ka + 1] * invd;
        b1v.x = W1n[ka * D_HID + col];
        b1v.y = W1n[(ka + 1) * D_HID + col];
        acc_s = __builtin_amdgcn_wmma_f32_16x16x4_f32(false, a0, false, b0,
                                                      (short)0, acc_s, false, false);
        acc_n = __builtin_amdgcn_wmma_f32_16x16x4_f32(false, a1, false, b1v,
                                                      (short)0, acc_n, false, false);
    }

    float bias = b1[col];
#pragma unroll
    for (int v = 0; v < 8; ++v) {
        int row = m * 16 + v + half * 8;           // C/D layout: VGPR v -> M = v (+8)
        float val = acc_s[v] + acc_n[v] + bias;
        h[row * D_HID + col] = fmaxf(val, 0.0f);
    }
}

// One wave per 16-node tile: D[16,16] = h_tile[16,256] @ Wpack[256,16],
// where Wpack cols 0-1 = W2_self, cols 2-3 = W2_neigh, cols 4-15 = 0.
// Split K into two accumulator chains for ILP.
__global__ void gemm2_wmma(const float* __restrict__ h,
                           const float* __restrict__ W2s,
                           const float* __restrict__ W2n,
                           float* __restrict__ out_s,   // self term -> d_out
                           float* __restrict__ y) {     // h @ W2_neigh (pre-scatter)
    int wave = (blockIdx.x * blockDim.x + threadIdx.x) >> 5;
    if (wave >= N_NODES / 16) return;    // wave-uniform: EXEC stays all-ones
    int lane = threadIdx.x & 31;
    int half = lane >> 4;
    int l15  = lane & 15;
    int rowA = wave * 16 + l15;
    const float* hr = h + rowA * D_HID;
    // lane's packed-B column: fixed base pointer (valid even for padding lanes)
    const float* wp = (l15 < 2) ? (W2s + (l15 & 1)) : (W2n + (l15 & 1));
    const bool active = (l15 < 4);

    v8f acc0 = {0.f,0.f,0.f,0.f,0.f,0.f,0.f,0.f};
    v8f acc1 = {0.f,0.f,0.f,0.f,0.f,0.f,0.f,0.f};

#pragma unroll 4
    for (int k = 0; k < D_HID / 2; k += 4) {
        int ka = k + half * 2;
        int kb = ka + D_HID / 2;
        v2f a0, b0, a1, b1v;
        a0.x = hr[ka];  a0.y = hr[ka + 1];
        a1.x = hr[kb];  a1.y = hr[kb + 1];
        b0.x = active ? wp[ka * D_OUT] : 0.0f;
        b0.y = active ? wp[(ka + 1) * D_OUT] : 0.0f;
        b1v.x = active ? wp[kb * D_OUT] : 0.0f;
        b1v.y = active ? wp[(kb + 1) * D_OUT] : 0.0f;
        acc0 = __builtin_amdgcn_wmma_f32_16x16x4_f32(false, a0, false, b0,
                                                     (short)0, acc0, false, false);
        acc1 = __builtin_amdgcn_wmma_f32_16x16x4_f32(false, a1, false, b1v,
                                                     (short)0, acc1, false, false);
    }

#pragma unroll
    for (int v = 0; v < 8; ++v) {
        int row = wave * 16 + v + half * 8;
        float val = acc0[v] + acc1[v];
        if (l15 < 2)       out_s[row * D_OUT + l15]       = val;
        else if (l15 < 4)  y[row * D_OUT + (l15 - 2)]     = val;
    }
}

// 2 floats/edge scatter of projected neighbor messages
__global__ void scatter2_kernel(const float* __restrict__ y,
                                const int* __restrict__ src,
                                const int* __restrict__ dst,
                                float* __restrict__ agg2) {
    int i = blockIdx.x * blockDim.x + threadIdx.x;   // < E*2
    if (i < N_EDGES * 2) {
        int e = i >> 1, c = i & 1;
        atomicAdd(&agg2[dst[e] * 2 + c], y[src[e] * 2 + c]);
    }
}

__global__ void finalize_kernel(float* __restrict__ out,
                                const float* __restrict__ agg2,
                                const float* __restrict__ deg,
                                const float* __restrict__ b2) {
    int i = blockIdx.x * blockDim.x + threadIdx.x;   // < N*2
    if (i < N_NODES * 2) {
        int n = i >> 1, c = i & 1;
        out[i] = out[i] + agg2[i] / fmaxf(deg[n], 1.0f) + b2[c];
    }
}

extern "C" void kernel_launch(void* const* d_in, const int* in_sizes, int n_in,
                              void* d_out, int out_size, void* d_ws, size_t ws_size,
                              hipStream_t stream) {
    const float* x   = (const float*)d_in[0];
    const int*   src = (const int*)  d_in[1];
    const int*   dst = (const int*)  d_in[2];
    const float* W1s = (const float*)d_in[3];
    const float* W1n = (const float*)d_in[4];
    const float* b1  = (const float*)d_in[5];
    const float* W2s = (const float*)d_in[6];
    const float* W2n = (const float*)d_in[7];
    const float* b2  = (const float*)d_in[8];
    float* out = (float*)d_out;
    float* ws  = (float*)d_ws;

    float* deg  = ws + OFF_DEG;
    float* agg1 = ws + OFF_AGG1;
    float* agg2 = ws + OFF_AGG2;
    float* h    = ws + OFF_H;
    float* y    = ws + OFF_Y;

    // 1) zero accumulation buffers (must happen every call; atomics accumulate)
    zero_f32<<<(ZERO_N + 255) / 256, 256, 0, stream>>>(ws, ZERO_N);
    // 2) degrees
    degree_kernel<<<(N_EDGES + 255) / 256, 256, 0, stream>>>(dst, deg);
    // 3) layer-1 edge aggregation of raw features (128 lanes per edge)
    scatter1_kernel<<<(N_EDGES * (D_IN / 256)) * 2, 256, 0, stream>>>(x, src, dst, agg1);
    // 4) layer-1 fused dual-GEMM + bias + relu (WMMA f32): 50000 waves, 4 waves/block
    gemm1_wmma<<<(N_NODES * 16 / 16) / 4, 128, 0, stream>>>(x, agg1, deg, W1s, W1n, b1, h);
    // 5) layer-2 packed GEMM (WMMA f32): self term -> out, neigh projection -> y
    gemm2_wmma<<<((N_NODES / 16) + 3) / 4, 128, 0, stream>>>(h, W2s, W2n, out, y);
    // 6) layer-2 edge aggregation of 2-d projected messages
    scatter2_kernel<<<(N_EDGES * 2 + 255) / 256, 256, 0, stream>>>(y, src, dst, agg2);
    // 7) out += agg2/deg + b2
    finalize_kernel<<<(N_NODES * 2 + 255) / 256, 256, 0, stream>>>(out, agg2, deg, b2);
}